// FactorPredictor_17918603559559
// MI455X (gfx1250) — compile-verified
//
#include <hip/hip_runtime.h>
#include <math.h>

typedef __attribute__((ext_vector_type(2))) float v2f;
typedef __attribute__((ext_vector_type(8))) float v8f;

#define HIDDEN   256
#define NFAC     64
#define BATCH    64
#define NSTK     8192
#define TILES_PER_B (NSTK / 16)          // 512
#define NTILES   (BATCH * TILES_PER_B)   // 32768
#define EPSN     1e-6f

// ---------------- stage 0: zero the 64x6 accumulator workspace ----------------
__global__ void fp_zero_acc(float* __restrict__ acc) {
    int i = threadIdx.x + blockIdx.x * blockDim.x;
    if (i < BATCH * 6) acc[i] = 0.0f;
}

// ---------------- stage 1: single streaming pass over e with WMMA -------------
// Per 16-stock tile: D = A(16x4) @ B(4x16) chained over K=256.
// B col0=q, col1=mu_w, col2=sig_w, cols 3..15 = 0.  Norm^2 via VALU FMAs.
__global__ __launch_bounds__(256)
void fp_pass1(const float* __restrict__ e,
              const float* __restrict__ q,
              const float* __restrict__ mu_w,
              const float* __restrict__ sig_w,
              float* __restrict__ acc)
{
    __shared__ float blds[4][HIDDEN];   // B source rows: q, mu_w, sig_w, zeros
    __shared__ float tbuf[8][48];       // per-wave transpose buffer [3][16]

    for (int idx = threadIdx.x; idx < 4 * HIDDEN; idx += 256) {
        int row = idx >> 8, k = idx & (HIDDEN - 1);
        float v = 0.0f;
        if (row == 0)      v = q[k];
        else if (row == 1) v = mu_w[k];
        else if (row == 2) v = sig_w[k];
        blds[row][k] = v;
    }
    __syncthreads();

    const int lane = threadIdx.x & 31;
    const int wib  = threadIdx.x >> 5;               // wave in block
    const int wid  = blockIdx.x * 8 + wib;           // global wave id
    const int nw   = gridDim.x * 8;                  // total waves (divides NTILES)
    const int col  = lane & 15;                      // D column / A row (M)
    const int kh   = lane >> 4;                      // K half select (0 or 1)
    const int brow = (col < 3) ? col : 3;            // B source row (3 => zeros)
    float* tw = &tbuf[wib][0];

    for (int t = wid; t < NTILES; t += nw) {
        const int b  = t >> 9;                        // batch
        const int n0 = (t & (TILES_PER_B - 1)) << 4;  // first stock of tile
        const float* ea = e + ((long)b * NSTK + n0 + col) * HIDDEN + 2 * kh;

        v8f d = {0.f, 0.f, 0.f, 0.f, 0.f, 0.f, 0.f, 0.f};
        float nrm = 0.0f;

        #pragma unroll 8
        for (int k0 = 0; k0 < HIDDEN; k0 += 4) {
            if ((k0 & 63) == 0)
                __builtin_prefetch(ea + k0 + 128, 0, 1);  // global_prefetch_b8
            v2f a  = *(const v2f*)(ea + k0);
            v2f bb = *(const v2f*)(&blds[brow][k0 + 2 * kh]);
            nrm = fmaf(a.x, a.x, nrm);
            nrm = fmaf(a.y, a.y, nrm);
            // D += A(16x4 f32) * B(4x16 f32)
            d = __builtin_amdgcn_wmma_f32_16x16x4_f32(
                    false, a, false, bb, (short)0, d, false, false);
        }
        // full row-norm^2: lane l and l^16 hold complementary K partials
        nrm += __shfl_xor(nrm, 16, 32);

        __syncthreads();
        if (col < 3) {   // lanes {0,1,2,16,17,18} own D columns 0..2
            int base = col * 16 + 8 * kh;            // rows base..base+7
            float4 lo = make_float4(d[0], d[1], d[2], d[3]);
            float4 hi = make_float4(d[4], d[5], d[6], d[7]);
            *(float4*)&tw[base]     = lo;
            *(float4*)&tw[base + 4] = hi;
        }
        __syncthreads();

        float s0 = 0.f, s1 = 0.f, s2 = 0.f, s3 = 0.f, s4 = 0.f, s5 = 0.f;
        if (lane < 16) {                              // lane j handles stock n0+j
            float dq  = tw[lane];
            float dm  = tw[16 + lane];
            float dsg = tw[32 + lane];
            float c   = dq / sqrtf(fmaxf(nrm, 1e-30f));   // q_norm folded in later
            float rp  = fmaxf(c, 0.0f);
            float rm  = fmaxf(-c, 0.0f);
            s0 = rp;  s1 = rp * dm;  s2 = rp * dsg;
            s3 = rm;  s4 = rm * dm;  s5 = rm * dsg;
        }
        #pragma unroll
        for (int off = 16; off > 0; off >>= 1) {
            s0 += __shfl_xor(s0, off, 32);
            s1 += __shfl_xor(s1, off, 32);
            s2 += __shfl_xor(s2, off, 32);
            s3 += __shfl_xor(s3, off, 32);
            s4 += __shfl_xor(s4, off, 32);
            s5 += __shfl_xor(s5, off, 32);
        }
        if (lane == 0) {
            float* ab = acc + b * 6;
            atomicAdd(ab + 0, s0);
            atomicAdd(ab + 1, s1);
            atomicAdd(ab + 2, s2);
            atomicAdd(ab + 3, s3);
            atomicAdd(ab + 4, s4);
            atomicAdd(ab + 5, s5);
        }
    }
}

// ---------------- stage 2: finalize 64x64 mu / sigma --------------------------
__global__ __launch_bounds__(256)
void fp_pass2(const float* __restrict__ acc,
              const float* __restrict__ w_key,
              const float* __restrict__ w_value,
              const float* __restrict__ q,
              const float* __restrict__ mu_b,
              const float* __restrict__ sig_b,
              float* __restrict__ out)
{
    __shared__ float red[256];
    float qv = q[threadIdx.x];                 // HIDDEN == blockDim == 256
    red[threadIdx.x] = qv * qv;
    __syncthreads();
    for (int s = 128; s > 0; s >>= 1) {
        if (threadIdx.x < s) red[threadIdx.x] += red[threadIdx.x + s];
        __syncthreads();
    }
    const float invqn = 1.0f / sqrtf(red[0]);
    const float mb = mu_b[0], sb = sig_b[0];

    for (int p = threadIdx.x; p < BATCH * NFAC; p += 256) {
        int b = p >> 6, k = p & (NFAC - 1);
        float wk = w_key[k];
        bool pos = (wk > 0.0f);
        const float* A = acc + b * 6;
        float S  = (pos ? A[0] : A[3]) * invqn;
        float MU = (pos ? A[1] : A[4]) * invqn;
        float SG = (pos ? A[2] : A[5]) * invqn;
        float inv = 1.0f / fmaxf(EPSN, S);
        float wv  = w_value[k];
        float mu  = fmaf(wv, MU * inv, mb);
        float x   = fmaf(wv, SG * inv, sb);
        float sp  = fmaxf(x, 0.0f) + log1pf(expf(-fabsf(x)));   // softplus
        out[p] = mu;
        out[BATCH * NFAC + p] = sp;
    }
}

extern "C" void kernel_launch(void* const* d_in, const int* in_sizes, int n_in,
                              void* d_out, int out_size, void* d_ws, size_t ws_size,
                              hipStream_t stream) {
    const float* e       = (const float*)d_in[0];
    const float* w_key   = (const float*)d_in[1];
    const float* w_value = (const float*)d_in[2];
    const float* q       = (const float*)d_in[3];
    const float* mu_w    = (const float*)d_in[4];
    const float* mu_b    = (const float*)d_in[5];
    const float* sig_w   = (const float*)d_in[6];
    const float* sig_b   = (const float*)d_in[7];
    float* out = (float*)d_out;
    float* acc = (float*)d_ws;                       // 64*6 floats scratch

    fp_zero_acc<<<1, 512, 0, stream>>>(acc);
    // 1024 blocks * 8 waves = 8192 waves; NTILES=32768 -> exactly 4 tiles/wave
    fp_pass1<<<1024, 256, 0, stream>>>(e, q, mu_w, sig_w, acc);
    fp_pass2<<<1, 256, 0, stream>>>(acc, w_key, w_value, q, mu_b, sig_b, out);
}